// SelfAttention_13881334301011
// MI455X (gfx1250) — compile-verified
//
#include <hip/hip_runtime.h>
#include <hip/hip_bf16.h>
#include <math.h>

// ---------------------------------------------------------------------------
// Problem constants (match reference)
// ---------------------------------------------------------------------------
#define T_DIM   4096
#define IN_DIM  1024
#define OUT_DIM 1024
#define KPAD    1040   // 1026 (OUT+2) padded up to a multiple of 16 for k-blocking
#define KB      16     // k-block staged through LDS in gemm_nn
#define EPS_F   1e-12f

typedef float v2f __attribute__((ext_vector_type(2)));
typedef float v8f __attribute__((ext_vector_type(8)));

// D = A(16x4,f32) * B(4x16,f32) + C(16x16,f32)  -- CDNA5 fp32 WMMA
__device__ __forceinline__ v8f wmma_f32(v2f a, v2f b, v8f c) {
  return __builtin_amdgcn_wmma_f32_16x16x4_f32(
      /*neg_a=*/false, a, /*neg_b=*/false, b,
      /*c_mod=*/(short)0, c, /*reuse_a=*/false, /*reuse_b=*/false);
}

// gfx1250 async global->LDS copy (inline asm: portable across both toolchains).
// Per lane: copies 4 bytes from the global address in gaddr to the LDS address in lds_off, tracked by ASYNCcnt.
__device__ __forceinline__ void async_ld_b32(unsigned lds_off, const float* gaddr) {
  asm volatile("global_load_async_to_lds_b32 %0, %1, off"
               :: "v"(lds_off), "v"(gaddr)
               : "memory");
}
__device__ __forceinline__ void wait_async0() {
  asm volatile("s_wait_asynccnt 0x0" ::: "memory");
}

// ---------------------------------------------------------------------------
// Block reduction helper (OP: 0=sum, 1=max, 2=product). Broadcasts result.
// ---------------------------------------------------------------------------
template <int OP>
__device__ __forceinline__ float blkred(float v, float* red) {
  const int tid = threadIdx.x;
  red[tid] = v;
  __syncthreads();
  for (int s = (int)blockDim.x >> 1; s > 0; s >>= 1) {
    if (tid < s) {
      float a = red[tid], b = red[tid + s];
      red[tid] = (OP == 0) ? (a + b) : (OP == 1) ? fmaxf(a, b) : (a * b);
    }
    __syncthreads();
  }
  float r = red[0];
  __syncthreads();
  return r;
}

// ---------------------------------------------------------------------------
// GEMM  C[M,N] = A[M,K] @ B[K,N]  (row-major, fp32 WMMA 16x16x4)
// block = 128 threads (4 waves); wave computes a 32x64 strip (8 accumulators).
// B is staged through LDS in transposed layout via async global->LDS loads so
// each B fragment read is one ds_load_b64.  grid = (N/64, M/128), K % 16 == 0.
// ---------------------------------------------------------------------------
__global__ __launch_bounds__(128)
void gemm_nn_wmma(const float* __restrict__ A, int lda,
                  const float* __restrict__ B, int ldb,
                  float* __restrict__ C, int ldc, int K) {
  __shared__ float sB[64 * 18];           // sB[c*18 + kk], kk in [0,16)
  const int tid  = threadIdx.x;
  const int lane = tid & 31;
  const int wave = tid >> 5;
  const int half = lane >> 4;
  const int l16  = lane & 15;
  const int r0 = (blockIdx.y * 4 + wave) * 32;
  const int c0 = blockIdx.x * 64;

  // async-copy assignment: thread -> (c, kk0..kk0+7), transposing into LDS
  const int lc  = tid & 63;
  const int lk0 = (tid >> 6) * 8;
  const unsigned lds0 = (unsigned)(uintptr_t)&sB[lc * 18 + lk0];

  const v8f vzero = {0, 0, 0, 0, 0, 0, 0, 0};
  v8f acc[2][4];
#pragma unroll
  for (int rt = 0; rt < 2; ++rt)
#pragma unroll
    for (int jt = 0; jt < 4; ++jt) acc[rt][jt] = vzero;

  const float* a0 = A + (size_t)(r0 + l16) * lda + 2 * half;
  const float* a1 = a0 + (size_t)16 * lda;

  for (int kb = 0; kb < K; kb += KB) {
    // stage B[kb..kb+16) x [c0..c0+64) into LDS, transposed
    const float* gb = B + (size_t)(kb + lk0) * ldb + c0 + lc;
#pragma unroll
    for (int u = 0; u < 8; ++u)
      async_ld_b32(lds0 + 4u * u, gb + (size_t)u * ldb);
    wait_async0();
    __syncthreads();

    if (kb + KB < K) __builtin_prefetch(a0 + kb + KB, 0, 3);  // global_prefetch_b8

#pragma unroll
    for (int ks = 0; ks < 4; ++ks) {
      const int k  = kb + 4 * ks;
      const int kk = 4 * ks + 2 * half;
      v2f aA = *(const v2f*)(a0 + k);
      v2f aB = *(const v2f*)(a1 + k);
#pragma unroll
      for (int jt = 0; jt < 4; ++jt) {
        v2f b = *(const v2f*)&sB[(jt * 16 + l16) * 18 + kk];  // ds_load_b64
        acc[0][jt] = wmma_f32(aA, b, acc[0][jt]);
        acc[1][jt] = wmma_f32(aB, b, acc[1][jt]);
      }
    }
    __syncthreads();
  }

#pragma unroll
  for (int rt = 0; rt < 2; ++rt)
#pragma unroll
    for (int v = 0; v < 8; ++v) {
      const int row = r0 + rt * 16 + v + 8 * half;
      float* crow = C + (size_t)row * ldc + c0 + l16;
      crow[0]  = acc[rt][0][v];
      crow[16] = acc[rt][1][v];
      crow[32] = acc[rt][2][v];
      crow[48] = acc[rt][3][v];
    }
}

// ---------------------------------------------------------------------------
// GEMM  C[M,N] = A[M,K] @ B[N,K]^T  (row-major). B fragments are contiguous
// per-lane float2, so direct global b64 loads are already efficient.
// Optional symmetric scale: C[i,j] *= scale[i]*scale[j] (cosine similarity).
// wave computes 32x64; grid = (N/64, M/128), K % 4 == 0.
// ---------------------------------------------------------------------------
__global__ __launch_bounds__(128)
void gemm_nt_wmma(const float* __restrict__ A, int lda,
                  const float* __restrict__ B, int ldb,
                  float* __restrict__ C, int ldc, int K,
                  const float* __restrict__ scale) {
  const int lane = threadIdx.x & 31;
  const int wave = threadIdx.x >> 5;
  const int half = lane >> 4;
  const int l16  = lane & 15;
  const int r0 = (blockIdx.y * 4 + wave) * 32;
  const int c0 = blockIdx.x * 64;

  const v8f vzero = {0, 0, 0, 0, 0, 0, 0, 0};
  v8f acc[2][4];
#pragma unroll
  for (int rt = 0; rt < 2; ++rt)
#pragma unroll
    for (int jt = 0; jt < 4; ++jt) acc[rt][jt] = vzero;

  const float* a0    = A + (size_t)(r0 + l16) * lda + 2 * half;
  const float* a1    = a0 + (size_t)16 * lda;
  const float* bbase = B + (size_t)(c0 + l16) * ldb + 2 * half;

  for (int k0 = 0; k0 < K; k0 += 4) {
    if (k0 + 64 < K) __builtin_prefetch(a0 + k0 + 64, 0, 3);
    v2f aA = *(const v2f*)(a0 + k0);
    v2f aB = *(const v2f*)(a1 + k0);
    v2f b0 = *(const v2f*)(bbase + k0);
    v2f b1 = *(const v2f*)(bbase + (size_t)16 * ldb + k0);
    v2f b2 = *(const v2f*)(bbase + (size_t)32 * ldb + k0);
    v2f b3 = *(const v2f*)(bbase + (size_t)48 * ldb + k0);
    acc[0][0] = wmma_f32(aA, b0, acc[0][0]);
    acc[1][0] = wmma_f32(aB, b0, acc[1][0]);
    acc[0][1] = wmma_f32(aA, b1, acc[0][1]);
    acc[1][1] = wmma_f32(aB, b1, acc[1][1]);
    acc[0][2] = wmma_f32(aA, b2, acc[0][2]);
    acc[1][2] = wmma_f32(aB, b2, acc[1][2]);
    acc[0][3] = wmma_f32(aA, b3, acc[0][3]);
    acc[1][3] = wmma_f32(aB, b3, acc[1][3]);
  }

  float cs0 = 1.f, cs1 = 1.f, cs2 = 1.f, cs3 = 1.f;
  if (scale) {
    cs0 = scale[c0 + l16];
    cs1 = scale[c0 + 16 + l16];
    cs2 = scale[c0 + 32 + l16];
    cs3 = scale[c0 + 48 + l16];
  }
#pragma unroll
  for (int rt = 0; rt < 2; ++rt)
#pragma unroll
    for (int v = 0; v < 8; ++v) {
      const int row = r0 + rt * 16 + v + 8 * half;
      const float rs = scale ? scale[row] : 1.0f;
      float* crow = C + (size_t)row * ldc + c0 + l16;
      crow[0]  = acc[rt][0][v] * rs * cs0;
      crow[16] = acc[rt][1][v] * rs * cs1;
      crow[32] = acc[rt][2][v] * rs * cs2;
      crow[48] = acc[rt][3][v] * rs * cs3;
    }
}

// ---------------------------------------------------------------------------
// 1/max(||x_i||, eps) per row
// ---------------------------------------------------------------------------
__global__ __launch_bounds__(256)
void rnorm_kernel(const float* __restrict__ x, float* __restrict__ rnorm) {
  __shared__ float red[256];
  const int i = blockIdx.x;
  float ss = 0.f;
  for (int c = threadIdx.x; c < IN_DIM; c += 256) {
    float v = x[(size_t)i * IN_DIM + c];
    ss += v * v;
  }
  ss = blkred<0>(ss, red);
  if (threadIdx.x == 0) rnorm[i] = 1.0f / fmaxf(sqrtf(ss), EPS_F);
}

// ---------------------------------------------------------------------------
// Per row of energies: softmax -> Att, entropy (raw, /log T), windowed entropy
// of energies*keep, and div_h = prod_j (1 - att_ij).
// ---------------------------------------------------------------------------
__global__ __launch_bounds__(256)
void softmax_entropy_kernel(const float* __restrict__ E,
                            float* __restrict__ Att,
                            float* __restrict__ ent_raw,
                            float* __restrict__ ent_win,
                            float* __restrict__ divh,
                            const int* __restrict__ bsp) {
  __shared__ float red[256];
  const int i   = blockIdx.x;
  const int tid = threadIdx.x;
  const int bs  = *bsp;
  const int blk = i / bs;
  const float* row = E + (size_t)i * T_DIM;
  const float inv_logT = 1.0f / logf((float)T_DIM);

  float m = -INFINITY, mv = -INFINITY;
  for (int j = tid; j < T_DIM; j += 256) {
    float e = row[j];
    m = fmaxf(m, e);
    float v = ((j / bs) == blk) ? e : 0.0f;
    mv = fmaxf(mv, v);
  }
  m  = blkred<1>(m,  red);
  mv = blkred<1>(mv, red);

  float s = 0.f, t1 = 0.f, s2 = 0.f, t2 = 0.f;
  for (int j = tid; j < T_DIM; j += 256) {
    float e  = row[j];
    float xx = e - m;
    float ex = expf(xx);
    s  += ex;
    t1 += xx * ex;
    float v  = ((j / bs) == blk) ? e : 0.0f;
    float x2 = v - mv;
    float e2 = expf(x2);
    s2 += e2;
    t2 += x2 * e2;
  }
  s  = blkred<0>(s,  red);
  t1 = blkred<0>(t1, red);
  s2 = blkred<0>(s2, red);
  t2 = blkred<0>(t2, red);

  if (tid == 0) {
    ent_raw[i] = (logf(s)  - t1 / s)  * inv_logT;   // H = log s - E[x e^x]/s
    ent_win[i] = (logf(s2) - t2 / s2) * inv_logT;
  }

  const float inv_s = 1.0f / s;
  float prod = 1.0f;
  for (int j = tid; j < T_DIM; j += 256) {
    float p = expf(row[j] - m) * inv_s;
    Att[(size_t)i * T_DIM + j] = p;
    prod *= (1.0f - p);
  }
  prod = blkred<2>(prod, red);
  if (tid == 0) divh[i] = prod;
}

// ---------------------------------------------------------------------------
// dep_factor[i] = sum_{j off-block} (1-sim)*att / sum_{j off-block} (1-sim)
// ---------------------------------------------------------------------------
__global__ __launch_bounds__(256)
void dep_kernel(const float* __restrict__ Sim, const float* __restrict__ Att,
                float* __restrict__ dep, const int* __restrict__ bsp) {
  __shared__ float red[256];
  const int i   = blockIdx.x;
  const int bs  = *bsp;
  const int blk = i / bs;
  float num = 0.f, den = 0.f;
  for (int j = threadIdx.x; j < T_DIM; j += 256) {
    if ((j / bs) != blk) {
      float d = 1.0f - Sim[(size_t)i * T_DIM + j];
      num += d * Att[(size_t)i * T_DIM + j];
      den += d;
    }
  }
  num = blkred<0>(num, red);
  den = blkred<0>(den, red);
  if (threadIdx.x == 0) dep[i] = num / den;   // reference divides without eps
}

// ---------------------------------------------------------------------------
// sums[0] = sum|ent_raw|, sums[1] = sum|div_h|
// ---------------------------------------------------------------------------
__global__ __launch_bounds__(1024)
void l1_sums_kernel(const float* __restrict__ ent_raw,
                    const float* __restrict__ divh,
                    float* __restrict__ sums) {
  __shared__ float red[1024];
  float a = 0.f, b = 0.f;
  for (int j = threadIdx.x; j < T_DIM; j += 1024) {
    a += fabsf(ent_raw[j]);
    b += fabsf(divh[j]);
  }
  a = blkred<0>(a, red);
  b = blkred<0>(b, red);
  if (threadIdx.x == 0) { sums[0] = a; sums[1] = b; }
}

// ---------------------------------------------------------------------------
// Write att_win, ent_win (broadcast), div_win (broadcast, L1-normalized)
// ---------------------------------------------------------------------------
__global__ __launch_bounds__(256)
void assemble_mats(const float* __restrict__ Att, const float* __restrict__ dep,
                   const float* __restrict__ entwin, const float* __restrict__ divh,
                   const float* __restrict__ sums, const int* __restrict__ bsp,
                   float* __restrict__ att_out, float* __restrict__ entw_out,
                   float* __restrict__ divw_out) {
  const int i = blockIdx.x;
  const int j = blockIdx.y * 256 + threadIdx.x;
  const int bs = *bsp;
  const bool keep = (i / bs) == (j / bs);
  const size_t idx = (size_t)i * T_DIM + j;
  att_out[idx]  = keep ? (Att[idx] + dep[j]) : 0.0f;
  entw_out[idx] = entwin[j];
  divw_out[idx] = divh[j] / sums[1];   // reference: no eps on div normalization
}

// ---------------------------------------------------------------------------
// concat[i, 0:1024]=ytmp, [1024]=entropy_norm, [1025]=dep, [1026:KPAD]=0
// ---------------------------------------------------------------------------
__global__ __launch_bounds__(256)
void build_concat(const float* __restrict__ ytmp, const float* __restrict__ ent_raw,
                  const float* __restrict__ dep, const float* __restrict__ sums,
                  float* __restrict__ cat) {
  const int i = blockIdx.x;
  for (int c = threadIdx.x; c < KPAD; c += 256) {
    float v;
    if (c < OUT_DIM)           v = ytmp[(size_t)i * OUT_DIM + c];
    else if (c == OUT_DIM)     v = ent_raw[i] / fmaxf(sums[0], EPS_F);
    else if (c == OUT_DIM + 1) v = dep[i];
    else                       v = 0.0f;
    cat[(size_t)i * KPAD + c] = v;
  }
}

// Zero-padded copy of Wout [1026,1024] -> [KPAD,1024]
__global__ __launch_bounds__(256)
void pad_wout(const float* __restrict__ Wout, float* __restrict__ Wp) {
  const int r = blockIdx.x;  // 0..KPAD-1
  for (int c = threadIdx.x; c < IN_DIM; c += 256) {
    Wp[(size_t)r * IN_DIM + c] =
        (r < OUT_DIM + 2) ? Wout[(size_t)r * IN_DIM + c] : 0.0f;
  }
}

// ---------------------------------------------------------------------------
// Host launch sequence (workspace need ~220 MB of floats)
// ---------------------------------------------------------------------------
extern "C" void kernel_launch(void* const* d_in, const int* in_sizes, int n_in,
                              void* d_out, int out_size, void* d_ws, size_t ws_size,
                              hipStream_t stream) {
  (void)in_sizes; (void)n_in; (void)out_size; (void)ws_size;

  const float* x    = (const float*)d_in[0];
  const float* Wk   = (const float*)d_in[1];
  const float* Wq   = (const float*)d_in[2];
  const float* Wv   = (const float*)d_in[3];
  const float* Wout = (const float*)d_in[4];
  const int*   bsp  = (const int*)d_in[5];

  float* out      = (float*)d_out;
  float* y_out    = out;                                     // [T, IN]
  float* att_out  = y_out  + (size_t)T_DIM * IN_DIM;         // [T, T]
  float* entw_out = att_out + (size_t)T_DIM * T_DIM;         // [T, T]
  float* divw_out = entw_out + (size_t)T_DIM * T_DIM;        // [T, T]

  float* w = (float*)d_ws;
  float* rnorm  = w; w += T_DIM;
  float* entraw = w; w += T_DIM;
  float* entwin = w; w += T_DIM;
  float* divh   = w; w += T_DIM;
  float* dep    = w; w += T_DIM;
  float* sums   = w; w += 16;
  float* Kbuf   = w; w += (size_t)T_DIM * OUT_DIM;
  float* Qbuf   = w; w += (size_t)T_DIM * OUT_DIM;
  float* Vbuf   = w; w += (size_t)T_DIM * OUT_DIM;
  float* Ebuf   = w; w += (size_t)T_DIM * T_DIM;   // energies, reused as similarity
  float* Attb   = w; w += (size_t)T_DIM * T_DIM;   // full (unmasked) softmax
  float* ytmp   = w; w += (size_t)T_DIM * OUT_DIM;
  float* catb   = w; w += (size_t)T_DIM * KPAD;
  float* Wp     = w; w += (size_t)KPAD * IN_DIM;

  const dim3 gBlk(128);
  const dim3 gProj(OUT_DIM / 64, T_DIM / 128);  // (16, 32)
  const dim3 gSq(T_DIM / 64, T_DIM / 128);      // (64, 32)

  // 1) row norms of x
  rnorm_kernel<<<T_DIM, 256, 0, stream>>>(x, rnorm);

  // 2) K = x@Wk, Q = x@Wq, V = x@Wv
  gemm_nn_wmma<<<gProj, gBlk, 0, stream>>>(x, IN_DIM, Wk, OUT_DIM, Kbuf, OUT_DIM, IN_DIM);
  gemm_nn_wmma<<<gProj, gBlk, 0, stream>>>(x, IN_DIM, Wq, OUT_DIM, Qbuf, OUT_DIM, IN_DIM);
  gemm_nn_wmma<<<gProj, gBlk, 0, stream>>>(x, IN_DIM, Wv, OUT_DIM, Vbuf, OUT_DIM, IN_DIM);

  // 3) energies = Q @ K^T
  gemm_nt_wmma<<<gSq, gBlk, 0, stream>>>(Qbuf, OUT_DIM, Kbuf, OUT_DIM, Ebuf, T_DIM,
                                         OUT_DIM, nullptr);

  // 4) softmax / entropies / prod(1-att)
  softmax_entropy_kernel<<<T_DIM, 256, 0, stream>>>(Ebuf, Attb, entraw, entwin, divh, bsp);

  // 5) similarity = (x@x^T) * rnorm_i * rnorm_j   (overwrites Ebuf)
  gemm_nt_wmma<<<gSq, gBlk, 0, stream>>>(x, IN_DIM, x, IN_DIM, Ebuf, T_DIM,
                                         IN_DIM, rnorm);

  // 6) dep_factor, L1 sums
  dep_kernel<<<T_DIM, 256, 0, stream>>>(Ebuf, Attb, dep, bsp);
  l1_sums_kernel<<<1, 1024, 0, stream>>>(entraw, divh, sums);

  // 7) assemble att_win / ent_win / div_win outputs
  assemble_mats<<<dim3(T_DIM, T_DIM / 256), 256, 0, stream>>>(
      Attb, dep, entwin, divh, sums, bsp, att_out, entw_out, divw_out);

  // 8) ytmp = att_win @ V  (att_win read from the output buffer)
  gemm_nn_wmma<<<gProj, gBlk, 0, stream>>>(att_out, T_DIM, Vbuf, OUT_DIM,
                                           ytmp, OUT_DIM, T_DIM);

  // 9) padded concat + padded Wout
  pad_wout<<<KPAD, 256, 0, stream>>>(Wout, Wp);
  build_concat<<<T_DIM, 256, 0, stream>>>(ytmp, entraw, dep, sums, catb);

  // 10) y = concat @ Wout
  gemm_nn_wmma<<<dim3(IN_DIM / 64, T_DIM / 128), gBlk, 0, stream>>>(
      catb, KPAD, Wp, IN_DIM, y_out, IN_DIM, KPAD);
}